// ResidualVectorQuantizer_28698971472093
// MI455X (gfx1250) — compile-verified
//
#include <hip/hip_runtime.h>

// Residual Vector Quantizer for MI455X (gfx1250, wave32, WMMA).
//
// Shapes: x [N=262144, D=32] f32, codebooks [4, 256, 32] f32.
// Outputs (concatenated flat in d_out, all as float):
//   x_q      [N*32]      at out[0 .. N*32)
//   mean_loss [1]        at out[N*32]
//   indices  [N*4]       at out[N*32+1 ...]   (float(idx), values 0..255 exact)
//
// Strategy: distances via V_WMMA_F32_16X16X4_F32 on scores
//   s[n,k] = dot(r_n, c_k) - 0.5*||c_k||^2   (argmin dist == argmax s)
// One wave owns a 16-row tile and runs all 4 stages sequentially from LDS.
// Column tiles are processed two-at-a-time with independent accumulators so
// the two WMMA RAW chains interleave on the matrix pipe.

typedef __attribute__((ext_vector_type(2))) float v2f;
typedef __attribute__((ext_vector_type(8))) float v8f;

#define WAVES     8          // waves per workgroup (256 threads)
#define DIM       32         // feature dim
#define KCB       256        // codes per stage
#define NSTAGE    4
#define CB_STRIDE 34         // padded LDS row stride (floats): conflict-free for
                             // 16-lane strided access (34*j distinct mod 64 for j<16)

// Branchless argmax select with jnp.argmin-compatible tie-break (lowest index).
__device__ __forceinline__ void takeMax(float& b, int& bi, float ob, int oi) {
    const bool take = (ob > b) | ((ob == b) & (oi < bi));
    b  = take ? ob : b;
    bi = take ? oi : bi;
}

__launch_bounds__(WAVES * 32)
__global__ void rvq_kernel(const float* __restrict__ x,
                           const float* __restrict__ cb,
                           float* __restrict__ out_xq,
                           float* __restrict__ out_idx,
                           float* __restrict__ ws_partial)
{
    __shared__ float cbL[NSTAGE * KCB * CB_STRIDE];  // 139264 B, padded codebooks
    __shared__ float hnL[NSTAGE * KCB];              //   4096 B, 0.5*||c||^2
    __shared__ float resL[WAVES * 16 * CB_STRIDE];   //  17408 B, per-wave residual tiles
    __shared__ float partL[WAVES];

    const int tid  = threadIdx.x;
    const int wave = tid >> 5;
    const int t    = tid & 31;
    const int j    = t & 15;      // lane-in-half: WMMA column / A-row index
    const int h    = t >> 4;      // half: selects K sub-pair (f32 x4 layout) / C row group

    // ---- cooperatively load codebooks into padded LDS + half-norms ----
    for (int e = tid; e < NSTAGE * KCB; e += WAVES * 32) {
        const float* src = cb + (size_t)e * DIM;     // 16B aligned in global
        float*       dst = cbL + e * CB_STRIDE;
        float hn = 0.f;
        #pragma unroll
        for (int f = 0; f < DIM; f += 4) {
            float4 v = *(const float4*)(src + f);
            dst[f + 0] = v.x; dst[f + 1] = v.y; dst[f + 2] = v.z; dst[f + 3] = v.w;
            hn += v.x * v.x + v.y * v.y + v.z * v.z + v.w * v.w;
        }
        hnL[e] = 0.5f * hn;
    }

    // ---- load this wave's 16x32 x-tile into its residual LDS slab ----
    const long  rowBase = (long)blockIdx.x * (WAVES * 16) + (long)wave * 16;
    float* resW = resL + wave * 16 * CB_STRIDE;
    {
        const float* src = x + (rowBase + j) * DIM + h * 16;   // lane: row j, half h
        float*       dst = resW + j * CB_STRIDE + h * 16;
        #pragma unroll
        for (int f = 0; f < 16; f += 2) {
            v2f v = *(const v2f*)(src + f);                    // 8B aligned
            *(v2f*)(dst + f) = v;                              // 136*j + 64*h + 4*f : 8B aligned
        }
    }
    __syncthreads();

    float lossAcc = 0.f;

    for (int qi = 0; qi < NSTAGE; ++qi) {
        // A fragments: A[m][k] = res[m][4c + k], lane supplies A[j][4c+2h .. 4c+2h+1]
        v2f a[8];
        #pragma unroll
        for (int c = 0; c < 8; ++c)
            a[c] = *(const v2f*)(resW + j * CB_STRIDE + 4 * c + 2 * h);

        float best[8];
        int   bidx[8];
        #pragma unroll
        for (int i = 0; i < 8; ++i) { best[i] = -__builtin_inff(); bidx[i] = 0; }

        const float* cbS = cbL + qi * KCB * CB_STRIDE;
        const float* hnS = hnL + qi * KCB;

        // Two column tiles per iteration -> two independent WMMA chains that
        // interleave on the matrix pipe instead of one serial RAW chain.
        for (int tile = 0; tile < 16; tile += 2) {
            const int   col0  = tile * 16 + j;
            const int   col1  = col0 + 16;
            const float bias0 = -hnS[col0];
            const float bias1 = -hnS[col1];
            v8f acc0, acc1;
            #pragma unroll
            for (int i = 0; i < 8; ++i) { acc0[i] = bias0; acc1[i] = bias1; }

            const float* b0p = cbS + col0 * CB_STRIDE + 2 * h;
            const float* b1p = cbS + col1 * CB_STRIDE + 2 * h;
            #pragma unroll
            for (int c = 0; c < 8; ++c) {
                // B[k][n] = cb[col][4c+k]; lane supplies B[2h..2h+1][col=j of tile]
                v2f b0 = *(const v2f*)(b0p + 4 * c);
                v2f b1 = *(const v2f*)(b1p + 4 * c);
                acc0 = __builtin_amdgcn_wmma_f32_16x16x4_f32(
                         false, a[c], false, b0, (short)0, acc0, false, false);
                acc1 = __builtin_amdgcn_wmma_f32_16x16x4_f32(
                         false, a[c], false, b1, (short)0, acc1, false, false);
            }

            #pragma unroll
            for (int i = 0; i < 8; ++i) {
                takeMax(best[i], bidx[i], acc0[i], col0);
                takeMax(best[i], bidx[i], acc1[i], col1);
            }
        }

        // cross-lane argmax over the 16 columns held by this half (branchless,
        // tie -> lowest index, matching jnp.argmin first-minimum semantics)
        #pragma unroll
        for (int mask = 1; mask < 16; mask <<= 1) {
            #pragma unroll
            for (int i = 0; i < 8; ++i) {
                const float ob = __shfl_xor(best[i], mask, 32);
                const int   oi = __shfl_xor(bidx[i], mask, 32);
                takeMax(best[i], bidx[i], ob, oi);
            }
        }

        // 16 active lanes: update residual row, emit index, accumulate loss.
        // Row m = j + 8h for j<8; (34m + f) mod 64 distinct across lanes -> conflict-free.
        if (j < 8) {
            const int m   = j + 8 * h;
            const int idx = bidx[j];
            out_idx[(rowBase + m) * NSTAGE + qi] = (float)idx;

            const float* cv = cbS + idx * CB_STRIDE;
            float*       rr = resW + m * CB_STRIDE;
            float ss = 0.f;
            #pragma unroll
            for (int f = 0; f < DIM; f += 2) {
                v2f r = *(v2f*)(rr + f);
                v2f c = *(const v2f*)(cv + f);
                r.x -= c.x; r.y -= c.y;
                *(v2f*)(rr + f) = r;
                ss += r.x * r.x + r.y * r.y;   // (q - r)^2 == new_residual^2
            }
            lossAcc += ss;
        }
        __syncthreads();   // re-converge; order LDS update vs next stage's fragment loads
    }

    // ---- x_q = x - final_residual ----
    {
        const float* xs = x      + (rowBase + j) * DIM + h * 16;
        const float* rs = resW   + j * CB_STRIDE + h * 16;
        float*       os = out_xq + (rowBase + j) * DIM + h * 16;
        #pragma unroll
        for (int f = 0; f < 16; f += 2) {
            v2f xv = *(const v2f*)(xs + f);
            v2f rv = *(const v2f*)(rs + f);
            v2f o; o.x = xv.x - rv.x; o.y = xv.y - rv.y;
            *(v2f*)(os + f) = o;
        }
    }

    // ---- deterministic loss partials: wave reduce -> LDS -> one value per block ----
    #pragma unroll
    for (int mask = 1; mask < 32; mask <<= 1)
        lossAcc += __shfl_xor(lossAcc, mask, 32);
    if (t == 0) partL[wave] = lossAcc;
    __syncthreads();
    if (tid == 0) {
        float s = 0.f;
        #pragma unroll
        for (int w = 0; w < WAVES; ++w) s += partL[w];
        ws_partial[blockIdx.x] = s;
    }
}

// Deterministic final reduction (fixed summation order; no float atomics).
__global__ void rvq_reduce(const float* __restrict__ partial, int nPart,
                           float* __restrict__ out_loss, float scale)
{
    __shared__ float sm[256];
    float s = 0.f;
    for (int i = threadIdx.x; i < nPart; i += 256) s += partial[i];
    sm[threadIdx.x] = s;
    __syncthreads();
    for (int w = 128; w > 0; w >>= 1) {
        if ((int)threadIdx.x < w) sm[threadIdx.x] += sm[threadIdx.x + w];
        __syncthreads();
    }
    if (threadIdx.x == 0) out_loss[0] = sm[0] * scale;
}

extern "C" void kernel_launch(void* const* d_in, const int* in_sizes, int n_in,
                              void* d_out, int out_size, void* d_ws, size_t ws_size,
                              hipStream_t stream)
{
    const float* x  = (const float*)d_in[0];   // [N, 32]
    const float* cb = (const float*)d_in[1];   // [4, 256, 32]
    const int N = in_sizes[0] / DIM;           // 262144

    float* out      = (float*)d_out;
    float* out_xq   = out;                          // [N*32]
    float* out_loss = out + (size_t)N * DIM;        // [1]
    float* out_idx  = out + (size_t)N * DIM + 1;    // [N*4] as float
    float* partial  = (float*)d_ws;

    const int blocks = N / (WAVES * 16);            // 2048

    rvq_kernel<<<blocks, WAVES * 32, 0, stream>>>(x, cb, out_xq, out_idx, partial);

    // mean over 4 stages of (1 + beta) * mean(residual^2) with beta = 0.25
    const float scale = 1.25f / (4.0f * (float)N * (float)DIM);
    rvq_reduce<<<1, 256, 0, stream>>>(partial, blocks, out_loss, scale);
}